// MGCN_42116449304721
// MI455X (gfx1250) — compile-verified
//
#include <hip/hip_runtime.h>

typedef __attribute__((ext_vector_type(2))) float v2f;
typedef __attribute__((ext_vector_type(4))) float f4;
typedef __attribute__((ext_vector_type(8))) float v8f;

// ---------- helpers ----------
__device__ __forceinline__ unsigned f2ord(float f) {
  unsigned u = __float_as_uint(f);
  return (u & 0x80000000u) ? ~u : (u | 0x80000000u);
}
__device__ __forceinline__ float ord2f(unsigned u) {
  unsigned v = (u & 0x80000000u) ? (u & 0x7FFFFFFFu) : ~u;
  return __uint_as_float(v);
}
#define ORD_NEG_INF 0x007FFFFFu  // f2ord(-inf)

__global__ void fill_f32(float* __restrict__ p, long long n, float v) {
  long long i = (long long)blockIdx.x * blockDim.x + threadIdx.x;
  if (i < n) p[i] = v;
}
__global__ void fill_u32(unsigned* __restrict__ p, long long n, unsigned v) {
  long long i = (long long)blockIdx.x * blockDim.x + threadIdx.x;
  if (i < n) p[i] = v;
}

// ---------- WMMA f32 GEMM: C[M,Nc] = A[rowIdx[m],K] @ B[K,Nc] ----------
// one wave per 16x16 C tile, V_WMMA_F32_16X16X4_F32, K-loop step 4
__global__ void wmma_gemm_f32(const float* __restrict__ A, const float* __restrict__ B,
                              float* __restrict__ C, const int* __restrict__ rowIdx,
                              int M, int K, int Nc) {
  const int lane = threadIdx.x & 31;
  const int wave = threadIdx.x >> 5;
  const int tilesN = Nc >> 4;
  const int tilesM = (M + 15) >> 4;
  int gw = blockIdx.x * (blockDim.x >> 5) + wave;
  if (gw >= tilesM * tilesN) return;
  const int tm = gw / tilesN, tn = gw % tilesN;
  const int row0 = tm << 4, col0 = tn << 4;
  const int half = lane >> 4, l16 = lane & 15;

  int ar = row0 + l16; if (ar > M - 1) ar = M - 1;
  if (rowIdx) ar = rowIdx[ar];
  const float* __restrict__ Arow = A + (size_t)ar * K;
  const float* __restrict__ Bcol = B + col0 + l16;

  v8f c = {0.f, 0.f, 0.f, 0.f, 0.f, 0.f, 0.f, 0.f};
  for (int kb = 0; kb < K; kb += 4) {
    const int ka = kb + (half << 1);          // lanes 0-15: K={kb,kb+1}; 16-31: K={kb+2,kb+3}
    v2f a, b;
    a.x = Arow[ka];
    a.y = Arow[ka + 1];
    b.x = Bcol[(size_t)ka * Nc];
    b.y = Bcol[(size_t)(ka + 1) * Nc];
    c = __builtin_amdgcn_wmma_f32_16x16x4_f32(false, a, false, b, (short)0, c, false, false);
  }
#pragma unroll
  for (int r = 0; r < 8; ++r) {
    int m = row0 + r + (half << 3);
    if (m < M) C[(size_t)m * Nc + col0 + l16] = c[r];
  }
}

// ---------- layer-1 edge attention (8 heads, 64 ch): wave per edge ----------
__global__ void edge_att8(const float* __restrict__ h, const int* __restrict__ src,
                          const int* __restrict__ dst, const int* __restrict__ rel,
                          const float* __restrict__ att, float* __restrict__ alpha,
                          unsigned* __restrict__ amax, int E, int Etot, int relSelf) {
  int e = blockIdx.x * (blockDim.x >> 5) + (threadIdx.x >> 5);
  if (e >= Etot) return;
  int lane = threadIdx.x & 31;
  int s, d, r;
  if (e < E) { s = src[e]; d = dst[e]; r = rel[e]; }
  else       { s = d = e - E; r = relSelf; }
  int hd = lane >> 2, part = lane & 3;       // 4 lanes per head, 16 ch each
  const float* hs = h + (size_t)s * 512 + hd * 64 + part * 16;
  const float* hi = h + (size_t)d * 512 + hd * 64 + part * 16;
  const float* at = att + ((size_t)r * 8 + hd) * 64 + part * 16;
  float acc = 0.f;
#pragma unroll
  for (int c = 0; c < 16; c += 4) {
    f4 a = *(const f4*)(hs + c);
    f4 b = *(const f4*)(hi + c);
    f4 w = *(const f4*)(at + c);
    acc += a.x * b.x * w.x + a.y * b.y * w.y + a.z * b.z * w.z + a.w * b.w * w.w;
  }
  acc += __shfl_xor(acc, 1, 32);
  acc += __shfl_xor(acc, 2, 32);
  if (part == 0) {
    float al = acc >= 0.f ? acc : 0.2f * acc;
    alpha[(size_t)e * 8 + hd] = al;
    atomicMax(&amax[(size_t)d * 8 + hd], f2ord(al));
  }
}

// ---------- exp + denom accumulate: thread per (edge, head) ----------
__global__ void edge_exp(float* __restrict__ alpha, const unsigned* __restrict__ amax,
                         float* __restrict__ denom, const int* __restrict__ dst,
                         int E, int Etot, int H) {
  long long i = (long long)blockIdx.x * blockDim.x + threadIdx.x;
  if (i >= (long long)Etot * H) return;
  int e = (int)(i / H), hd = (int)(i % H);
  int d = (e < E) ? dst[e] : e - E;
  float m = ord2f(amax[(size_t)d * H + hd]);
  float ea = __expf(alpha[i] - m);
  alpha[i] = ea;
  atomicAdd(&denom[(size_t)d * H + hd], ea);
}

// ---------- layer-1 message scatter: wave per edge ----------
__global__ void edge_msg8(const float* __restrict__ h, const float* __restrict__ alpha,
                          const float* __restrict__ denom, const int* __restrict__ src,
                          const int* __restrict__ dst, float* __restrict__ out,
                          int E, int Etot) {
  int e = blockIdx.x * (blockDim.x >> 5) + (threadIdx.x >> 5);
  if (e >= Etot) return;
  int lane = threadIdx.x & 31;
  int s, d;
  if (e < E) { s = src[e]; d = dst[e]; } else { s = d = e - E; }
  int hd = lane >> 2, part = lane & 3;
  float w = alpha[(size_t)e * 8 + hd] / denom[(size_t)d * 8 + hd];
  const float* hs = h + (size_t)s * 512 + hd * 64 + part * 16;
  float* od = out + (size_t)d * 512 + hd * 64 + part * 16;
#pragma unroll
  for (int c = 0; c < 16; ++c) atomicAdd(&od[c], hs[c] * w);
}

// ---------- bias + leaky-relu (slope=1 -> pure bias add) ----------
__global__ void bias_act(float* __restrict__ x, const float* __restrict__ b,
                         int cols, long long total, float slope) {
  long long i = (long long)blockIdx.x * blockDim.x + threadIdx.x;
  if (i >= total) return;
  int j = (int)(i % cols);
  float v = x[i] + b[j];
  x[i] = v >= 0.f ? v : slope * v;
}

// ---------- layer-2 edge attention (1 head, 64 ch): wave per edge ----------
__global__ void edge_att1(const float* __restrict__ h, const int* __restrict__ src,
                          const int* __restrict__ dst, const int* __restrict__ rel,
                          const float* __restrict__ att, float* __restrict__ alpha,
                          unsigned* __restrict__ amax, int E, int Etot, int relSelf) {
  int e = blockIdx.x * (blockDim.x >> 5) + (threadIdx.x >> 5);
  if (e >= Etot) return;
  int lane = threadIdx.x & 31;
  int s, d, r;
  if (e < E) { s = src[e]; d = dst[e]; r = rel[e]; }
  else       { s = d = e - E; r = relSelf; }
  const float* hs = h + (size_t)s * 64;
  const float* hi = h + (size_t)d * 64;
  const float* at = att + (size_t)r * 64;
  float acc = hs[lane] * hi[lane] * at[lane] + hs[lane + 32] * hi[lane + 32] * at[lane + 32];
#pragma unroll
  for (int m = 1; m < 32; m <<= 1) acc += __shfl_xor(acc, m, 32);
  if (lane == 0) {
    float al = acc >= 0.f ? acc : 0.2f * acc;
    alpha[e] = al;
    atomicMax(&amax[d], f2ord(al));
  }
}

// ---------- layer-2 message scatter ----------
__global__ void edge_msg1(const float* __restrict__ h, const float* __restrict__ alpha,
                          const float* __restrict__ denom, const int* __restrict__ src,
                          const int* __restrict__ dst, float* __restrict__ out,
                          int E, int Etot) {
  int e = blockIdx.x * (blockDim.x >> 5) + (threadIdx.x >> 5);
  if (e >= Etot) return;
  int lane = threadIdx.x & 31;
  int s, d;
  if (e < E) { s = src[e]; d = dst[e]; } else { s = d = e - E; }
  float w = alpha[e] / denom[d];
  atomicAdd(&out[(size_t)d * 64 + lane], h[(size_t)s * 64 + lane] * w);
  atomicAdd(&out[(size_t)d * 64 + lane + 32], h[(size_t)s * 64 + lane + 32] * w);
}

extern "C" void kernel_launch(void* const* d_in, const int* in_sizes, int n_in,
                              void* d_out, int out_size, void* d_ws, size_t ws_size,
                              hipStream_t stream) {
  const int*   entity    = (const int*)d_in[0];
  const int*   edge_idx  = (const int*)d_in[1];
  const int*   edge_attr = (const int*)d_in[2];
  const float* ent_emb   = (const float*)d_in[3];
  const float* W1        = (const float*)d_in[4];
  const float* att1      = (const float*)d_in[5];
  const float* b1        = (const float*)d_in[6];
  const float* W2        = (const float*)d_in[7];
  const float* att2      = (const float*)d_in[8];
  const float* b2        = (const float*)d_in[9];

  const int N = in_sizes[0];
  const int E = in_sizes[1] / 2;
  const int D = in_sizes[3] / N;            // 64
  const int H1D = in_sizes[6];              // 512
  const int H1 = H1D / D;                   // 8
  const int relSelf = in_sizes[5] / (H1 * D) - 1;  // 400
  const int Etot = E + N;
  const int* src = edge_idx;
  const int* dst = edge_idx + E;

  float* ws = (float*)d_ws;
  size_t off = 0;
  float* h1     = ws + off; off += (size_t)N * H1D;     // also reused as h2
  float* out1   = ws + off; off += (size_t)N * H1D;     // also x2 after bias/leaky
  float* alpha1 = ws + off; off += (size_t)Etot * H1;   // also alpha2
  unsigned* amax1 = (unsigned*)(ws + off); off += (size_t)N * H1;  // also amax2
  float* denom1 = ws + off; off += (size_t)N * H1;      // also denom2
  float* h2 = h1;
  float* alpha2 = alpha1;
  unsigned* amax2 = amax1;
  float* denom2 = denom1;
  float* out = (float*)d_out;

  const int TB = 256;
  auto cdiv = [](long long a, long long b) { return (int)((a + b - 1) / b); };

  // ---- layer 1 ----
  {
    int tiles = ((N + 15) / 16) * (H1D / 16);
    wmma_gemm_f32<<<cdiv(tiles, TB / 32), TB, 0, stream>>>(ent_emb, W1, h1, entity, N, D, H1D);
  }
  fill_f32<<<cdiv((long long)N * H1D, TB), TB, 0, stream>>>(out1, (long long)N * H1D, 0.f);
  fill_u32<<<cdiv((long long)N * H1, TB), TB, 0, stream>>>(amax1, (long long)N * H1, ORD_NEG_INF);
  fill_f32<<<cdiv((long long)N * H1, TB), TB, 0, stream>>>(denom1, (long long)N * H1, 0.f);

  edge_att8<<<cdiv(Etot, TB / 32), TB, 0, stream>>>(h1, src, dst, edge_attr, att1, alpha1, amax1,
                                                    E, Etot, relSelf);
  edge_exp<<<cdiv((long long)Etot * H1, TB), TB, 0, stream>>>(alpha1, amax1, denom1, dst,
                                                              E, Etot, H1);
  edge_msg8<<<cdiv(Etot, TB / 32), TB, 0, stream>>>(h1, alpha1, denom1, src, dst, out1, E, Etot);
  bias_act<<<cdiv((long long)N * H1D, TB), TB, 0, stream>>>(out1, b1, H1D, (long long)N * H1D,
                                                            0.01f);

  // ---- layer 2 ----
  {
    int tiles = ((N + 15) / 16) * (D / 16);
    wmma_gemm_f32<<<cdiv(tiles, TB / 32), TB, 0, stream>>>(out1, W2, h2, nullptr, N, H1D, D);
  }
  fill_u32<<<cdiv(N, TB), TB, 0, stream>>>(amax2, N, ORD_NEG_INF);
  fill_f32<<<cdiv(N, TB), TB, 0, stream>>>(denom2, N, 0.f);
  fill_f32<<<cdiv((long long)N * D, TB), TB, 0, stream>>>(out, (long long)N * D, 0.f);

  edge_att1<<<cdiv(Etot, TB / 32), TB, 0, stream>>>(h2, src, dst, edge_attr, att2, alpha2, amax2,
                                                    E, Etot, relSelf);
  edge_exp<<<cdiv(Etot, TB), TB, 0, stream>>>(alpha2, amax2, denom2, dst, E, Etot, 1);
  edge_msg1<<<cdiv(Etot, TB / 32), TB, 0, stream>>>(h2, alpha2, denom2, src, dst, out, E, Etot);
  bias_act<<<cdiv((long long)N * D, TB), TB, 0, stream>>>(out, b2, D, (long long)N * D, 1.0f);
}